// Net_15625091023093
// MI455X (gfx1250) — compile-verified
//
#include <hip/hip_runtime.h>
#include <math.h>

// ---------------------------------------------------------------------------
// 3-layer weighted GraphConv (rel+root) + linear head + log_softmax
// N=100000, E=1600000, F_IN=100, HID=128, CLS=47
//
// GEMM dataflow: W^T fragments live in registers for the whole kernel
// (64 VGPRs/wave), X tiles are staged via LDS (async-to-LDS when available),
// inner loop is a pure v_wmma_f32_16x16x4_f32 accumulate chain.
// ---------------------------------------------------------------------------

typedef float v2f __attribute__((ext_vector_type(2)));
typedef float v8f __attribute__((ext_vector_type(8)));
typedef int   v4i __attribute__((ext_vector_type(4)));

#define NN 100000
#define EE 1600000
#define HID 128
#define CLS 47

// ---- CDNA5 async global->LDS staging (guarded; falls back to float4 copy) --
#if defined(__has_builtin)
#if __has_builtin(__builtin_amdgcn_global_load_async_to_lds_b128) && \
    __has_builtin(__builtin_amdgcn_s_wait_asynccnt)
#define USE_ASYNC_LDS 1
#endif
#endif

typedef __attribute__((address_space(1))) v4i* g_v4i_ptr;   // global (device)
typedef __attribute__((address_space(3))) v4i* l_v4i_ptr;   // LDS (shared)

__device__ __forceinline__ void g2l_16B(const float* g, float* l) {
#ifdef USE_ASYNC_LDS
    __builtin_amdgcn_global_load_async_to_lds_b128(
        (g_v4i_ptr)g, (l_v4i_ptr)l, 0, 0);
#else
    *(float4*)l = *(const float4*)g;
#endif
}

__device__ __forceinline__ void g2l_wait() {
#ifdef USE_ASYNC_LDS
    __builtin_amdgcn_s_wait_asynccnt(0);
#endif
}

// ---------------------------------------------------------------------------
// C[M x 128] = X[M x K] * W^T   (W row-major [128 x K]); optional +bias.
// Block = 256 threads (8 waves). Wave w owns output columns [16w, 16w+16):
// its B fragments (K/4 x v2f) are loaded ONCE into registers, then the block
// grid-strides over 16-row tiles of X staged through LDS.
// A-frag (16x4 f32): lanes 0-15 K={k,k+1}, lanes 16-31 K={k+2,k+3}.
// D (16x16 f32): lane n = lane&15; VGPR r holds M = r + 8*(lane>>4).
// ---------------------------------------------------------------------------
template <int K>
__global__ void gemm_xwt_128(const float* __restrict__ X,
                             const float* __restrict__ W,
                             const float* __restrict__ bias,
                             float* __restrict__ C, int M) {
    constexpr int K4 = K / 4;
    constexpr int SP = K + 4;                 // padded LDS row stride (floats)
    __shared__ float sX[16 * SP];

    const int tid  = threadIdx.x;
    const int lane = tid & 31;
    const int wv   = tid >> 5;                // 0..7 -> column tile
    const int row  = lane & 15;
    const int kh   = lane >> 4;               // K-half selector
    const int n    = wv * 16 + row;           // output column

    // ---- B tile (W^T columns) -> registers, once per block ----
    v2f breg[K4];
#pragma unroll
    for (int ks = 0; ks < K4; ++ks)
        breg[ks] = *(const v2f*)(W + (size_t)n * K + ks * 4 + kh * 2);

    const float bv = bias ? bias[n] : 0.0f;
    const int numTiles = M / 16;

    for (int tile = blockIdx.x; tile < numTiles; tile += gridDim.x) {
        const int m0 = tile * 16;

        // cooperative coalesced/async fill of the 16xK X tile
        for (int idx = tid; idx < 16 * K4; idx += 256) {
            const int r  = idx / K4;
            const int c4 = idx - r * K4;
            g2l_16B(X + (size_t)(m0 + r) * K + c4 * 4, &sX[r * SP + c4 * 4]);
        }
        // prefetch next tile while this one is consumed
        const int tnext = tile + gridDim.x;
        if (tnext < numTiles && tid < 64) {
            const int pr = tid >> 2, seg = tid & 3;
            __builtin_prefetch(X + (size_t)(tnext * 16 + pr) * K + seg * 32, 0, 3);
        }
        g2l_wait();
        __syncthreads();

        // A fragments -> registers in one DS burst
        v2f a[K4];
#pragma unroll
        for (int ks = 0; ks < K4; ++ks)
            a[ks] = *(const v2f*)(&sX[row * SP + ks * 4 + kh * 2]);
        __syncthreads();                      // LDS free for next tile

        // pure WMMA accumulate chain
        v8f acc = {};
#pragma unroll
        for (int ks = 0; ks < K4; ++ks)
            acc = __builtin_amdgcn_wmma_f32_16x16x4_f32(
                false, a[ks], false, breg[ks], (short)0, acc, false, false);

#pragma unroll
        for (int r = 0; r < 8; ++r)
            C[(size_t)(m0 + r + 8 * kh) * HID + n] = acc[r] + bv;
    }
}

// ---------------------------------------------------------------------------
// logits[M x 47] (+)= Xk[M x 128] @ Wl^T ; Wl = W_lin + 128*layer, ldw = 384.
// Block = 96 threads (3 waves), wave w owns columns [16w, 16w+16), masked to
// n<47 on loads/stores only (EXEC all-ones for WMMA). first!=0 seeds bias.
// ---------------------------------------------------------------------------
__global__ void classifier_accum(const float* __restrict__ Xk,
                                 const float* __restrict__ Wl,
                                 const float* __restrict__ bias,
                                 float* __restrict__ logits,
                                 int M, int first) {
    constexpr int K = HID, K4 = K / 4, SP = K + 4;
    __shared__ float sX[16 * SP];

    const int tid  = threadIdx.x;
    const int lane = tid & 31;
    const int wv   = tid >> 5;                // 0..2
    const int row  = lane & 15;
    const int kh   = lane >> 4;
    const int n    = wv * 16 + row;
    const bool valid = (n < CLS);

    v2f breg[K4];
#pragma unroll
    for (int ks = 0; ks < K4; ++ks) {
        if (valid) breg[ks] = *(const v2f*)(Wl + (size_t)n * (3 * HID) + ks * 4 + kh * 2);
        else       { breg[ks].x = 0.0f; breg[ks].y = 0.0f; }
    }
    const float bv = valid ? bias[n] : 0.0f;
    const int numTiles = M / 16;

    for (int tile = blockIdx.x; tile < numTiles; tile += gridDim.x) {
        const int m0 = tile * 16;
        for (int idx = tid; idx < 16 * K4; idx += 96) {
            const int r  = idx / K4;
            const int c4 = idx - r * K4;
            g2l_16B(Xk + (size_t)(m0 + r) * K + c4 * 4, &sX[r * SP + c4 * 4]);
        }
        g2l_wait();
        __syncthreads();

        v2f a[K4];
#pragma unroll
        for (int ks = 0; ks < K4; ++ks)
            a[ks] = *(const v2f*)(&sX[row * SP + ks * 4 + kh * 2]);
        __syncthreads();

        v8f acc = {};
#pragma unroll
        for (int ks = 0; ks < K4; ++ks)
            acc = __builtin_amdgcn_wmma_f32_16x16x4_f32(
                false, a[ks], false, breg[ks], (short)0, acc, false, false);

        if (valid) {
#pragma unroll
            for (int r = 0; r < 8; ++r) {
                const size_t off = (size_t)(m0 + r + 8 * kh) * CLS + n;
                const float prev = first ? bv : logits[off];
                logits[off] = prev + acc[r];
            }
        }
    }
}

// ---------------------------------------------------------------------------
// acc[dst] += ew * xr[src] ; one 32-lane group per edge, 4 floats per lane
// (512B coalesced gather per edge, f32 atomics into L2-resident acc).
// ---------------------------------------------------------------------------
__global__ void edge_scatter(const int* __restrict__ src,
                             const int* __restrict__ dst,
                             const float* __restrict__ ew,
                             const float* __restrict__ xr,
                             float* __restrict__ acc) {
    const long long gid = (long long)blockIdx.x * blockDim.x + threadIdx.x;
    const long long e = gid >> 5;
    if (e >= EE) return;
    const int c = (int)(gid & 31) * 4;

    const int   s = src[e];
    const int   d = dst[e];
    const float w = ew[e];

    const float4 v = *(const float4*)(xr + (size_t)s * HID + c);
    float* p = acc + (size_t)d * HID + c;
    atomicAdd(p + 0, v.x * w);
    atomicAdd(p + 1, v.y * w);
    atomicAdd(p + 2, v.z * w);
    atomicAdd(p + 3, v.w * w);
}

// ---------------------------------------------------------------------------
__global__ void relu_inplace(float* __restrict__ x, long long n4) {
    const long long i = (long long)blockIdx.x * blockDim.x + threadIdx.x;
    if (i >= n4) return;
    float4 v = ((float4*)x)[i];
    v.x = fmaxf(v.x, 0.0f); v.y = fmaxf(v.y, 0.0f);
    v.z = fmaxf(v.z, 0.0f); v.w = fmaxf(v.w, 0.0f);
    ((float4*)x)[i] = v;
}

// ---------------------------------------------------------------------------
__global__ void log_softmax_47(const float* __restrict__ logits,
                               float* __restrict__ out, int M) {
    const int i = blockIdx.x * blockDim.x + threadIdx.x;
    if (i >= M) return;
    const float* l = logits + (size_t)i * CLS;
    float mx = -INFINITY;
    for (int j = 0; j < CLS; ++j) mx = fmaxf(mx, l[j]);
    float s = 0.0f;
    for (int j = 0; j < CLS; ++j) s += expf(l[j] - mx);
    const float lse = logf(s) + mx;
    float* o = out + (size_t)i * CLS;
    for (int j = 0; j < CLS; ++j) o[j] = l[j] - lse;
}

// ---------------------------------------------------------------------------
extern "C" void kernel_launch(void* const* d_in, const int* in_sizes, int n_in,
                              void* d_out, int out_size, void* d_ws, size_t ws_size,
                              hipStream_t stream) {
    const float* x0      = (const float*)d_in[0];      // [N,100]
    const int*   ei      = (const int*)  d_in[1];      // [2,E]
    const float* ew      = (const float*)d_in[2];      // [E]
    const float* W1_rel  = (const float*)d_in[3];      // [128,100]
    const float* W1_root = (const float*)d_in[4];
    const float* b1      = (const float*)d_in[5];
    const float* W2_rel  = (const float*)d_in[6];      // [128,128]
    const float* W2_root = (const float*)d_in[7];
    const float* b2      = (const float*)d_in[8];
    const float* W3_rel  = (const float*)d_in[9];
    const float* W3_root = (const float*)d_in[10];
    const float* b3      = (const float*)d_in[11];
    const float* W_lin   = (const float*)d_in[12];     // [47,384]
    const float* b_lin   = (const float*)d_in[13];

    const int* src = ei;
    const int* dst = ei + EE;

    float* ws     = (float*)d_ws;
    float* xr     = ws;                                // [N,128] rel-GEMM scratch
    float* bufA   = ws + (size_t)NN * HID;             // [N,128] x1 / x3
    float* bufB   = ws + (size_t)2 * NN * HID;         // [N,128] x2
    float* logits = ws + (size_t)3 * NN * HID;         // [N,47]

    const dim3 gB(1250), gT(256);                      // 6250 row tiles, 5/block
    const dim3 cB(2084), cT(96);                       // classifier: 3 waves/blk
    const long long scat = (long long)EE * 32;
    const int scatBlocks = (int)((scat + 255) / 256);
    const long long n4 = (long long)NN * HID / 4;
    const int reluBlocks = (int)((n4 + 255) / 256);
    const int smBlocks = (NN + 255) / 256;

    // ---- Layer 1 (K = 100) ----
    gemm_xwt_128<100><<<gB, gT, 0, stream>>>(x0, W1_rel,  nullptr, xr,   NN);
    gemm_xwt_128<100><<<gB, gT, 0, stream>>>(x0, W1_root, b1,      bufA, NN);
    edge_scatter<<<scatBlocks, 256, 0, stream>>>(src, dst, ew, xr, bufA);
    relu_inplace<<<reluBlocks, 256, 0, stream>>>(bufA, n4);
    classifier_accum<<<cB, cT, 0, stream>>>(bufA, W_lin,       b_lin, logits, NN, 1);

    // ---- Layer 2 (K = 128) ----
    gemm_xwt_128<128><<<gB, gT, 0, stream>>>(bufA, W2_rel,  nullptr, xr,   NN);
    gemm_xwt_128<128><<<gB, gT, 0, stream>>>(bufA, W2_root, b2,      bufB, NN);
    edge_scatter<<<scatBlocks, 256, 0, stream>>>(src, dst, ew, xr, bufB);
    relu_inplace<<<reluBlocks, 256, 0, stream>>>(bufB, n4);
    classifier_accum<<<cB, cT, 0, stream>>>(bufB, W_lin + 128, b_lin, logits, NN, 0);

    // ---- Layer 3 (K = 128, reuse bufA) ----
    gemm_xwt_128<128><<<gB, gT, 0, stream>>>(bufB, W3_rel,  nullptr, xr,   NN);
    gemm_xwt_128<128><<<gB, gT, 0, stream>>>(bufB, W3_root, b3,      bufA, NN);
    edge_scatter<<<scatBlocks, 256, 0, stream>>>(src, dst, ew, xr, bufA);
    relu_inplace<<<reluBlocks, 256, 0, stream>>>(bufA, n4);
    classifier_accum<<<cB, cT, 0, stream>>>(bufA, W_lin + 256, b_lin, logits, NN, 0);

    // ---- Head ----
    log_softmax_47<<<smBlocks, 256, 0, stream>>>(logits, (float*)d_out, NN);
}